// Decoder_Attention_55327768707598
// MI455X (gfx1250) — compile-verified
//
#include <hip/hip_runtime.h>
#include <math.h>

// Problem constants (match reference).
#define BB    32
#define TDEC  64
#define SENC  2048
#define HH    256
#define EE    256

typedef __attribute__((ext_vector_type(2))) float v2f;
typedef __attribute__((ext_vector_type(8))) float v8f;

// gfx1250 has a hardware v_tanh_f32 (TRANS op). Use it if the builtin exists.
#if __has_builtin(__builtin_amdgcn_tanhf)
__device__ __forceinline__ float fast_tanh(float x) { return __builtin_amdgcn_tanhf(x); }
#else
__device__ __forceinline__ float fast_tanh(float x) { return tanhf(x); }
#endif

// ---------------------------------------------------------------------------
// enc_proj[b*S+s][n] = sum_k enc_hiddens[b*S+s][k] * Wk[k][n]
// Wk = W rows 256..511.  M = B*S = 65536, N = 256, K = 256, all fp32.
// fp32 WMMA 16x16x4:
//   A frag (2 VGPR): lane<16 -> row M=tileM+lane, {K,K+1}; lane>=16 -> {K+2,K+3}
//   B frag (2 VGPR): lane<16 -> col N=tileN+lane, rows {K,K+1}; lane>=16 -> {K+2,K+3}
//   C frag (8 VGPR): element r -> row tileM + 8*(lane/16) + r, col tileN + lane%16
// One wave owns two N-tiles (A fragment reused); block = 8 waves = 16x256 slab.
// ---------------------------------------------------------------------------
__global__ __launch_bounds__(256) void k_enc_proj(const float* __restrict__ A,
                                                  const float* __restrict__ W,
                                                  float* __restrict__ C)
{
    const float* Bk = W + 256 * 256; // Wk
    const int wave = threadIdx.x >> 5;
    const int lane = threadIdx.x & 31;
    const int half = lane >> 4;   // 0: K,K+1   1: K+2,K+3
    const int lr   = lane & 15;
    const int tileM = blockIdx.x * 16;
    const int n0 = wave * 32;
    const int n1 = n0 + 16;

    const float* arow = A  + (size_t)(tileM + lr) * 256 + half * 2;
    const float* b0p  = Bk + (size_t)(half * 2) * 256 + n0 + lr;
    const float* b1p  = Bk + (size_t)(half * 2) * 256 + n1 + lr;

    v8f c0 = {}; v8f c1 = {};
    for (int k = 0; k < 256; k += 4) {
        v2f a;  a.x  = arow[0]; a.y  = arow[1];
        v2f b0; b0.x = b0p[0];  b0.y = b0p[256];
        v2f b1; b1.x = b1p[0];  b1.y = b1p[256];
        c0 = __builtin_amdgcn_wmma_f32_16x16x4_f32(false, a, false, b0, (short)0, c0, false, false);
        c1 = __builtin_amdgcn_wmma_f32_16x16x4_f32(false, a, false, b1, (short)0, c1, false, false);
        arow += 4;
        b0p  += 4 * 256;
        b1p  += 4 * 256;
    }

    float* crow = C + (size_t)(tileM + half * 8) * 256;
#pragma unroll
    for (int r = 0; r < 8; ++r) {
        crow[(size_t)r * 256 + n0 + lr] = c0[r];
        crow[(size_t)r * 256 + n1 + lr] = c1[r];
    }
}

// ---------------------------------------------------------------------------
// hidden = last_hidden (copy to ws), q = last_hidden @ Wq  (Wq = W rows 0..255)
// One block per batch row; hidden row cached in LDS.
// ---------------------------------------------------------------------------
__global__ __launch_bounds__(256) void k_init(const float* __restrict__ last_hidden,
                                              const float* __restrict__ W,
                                              float* __restrict__ hidden,
                                              float* __restrict__ q)
{
    __shared__ float hl[256];
    const int b = blockIdx.x, h = threadIdx.x;
    const float v = last_hidden[b * 256 + h];
    hidden[b * 256 + h] = v;
    hl[h] = v;
    __syncthreads();
    float acc = 0.f;
#pragma unroll 4
    for (int k = 0; k < 256; ++k) acc += hl[k] * W[k * 256 + h];
    q[b * 256 + h] = acc;
}

// ---------------------------------------------------------------------------
// scores[row] = dot(tanh(q[b] + enc_proj[row]), V), row = b*S+s.
// One wave32 per row; 8 contiguous floats per lane; shfl_xor wave reduction.
// Streams 64MB of enc_proj per step — resident in the 192MB L2.
// ---------------------------------------------------------------------------
__global__ __launch_bounds__(256) void k_scores(const float* __restrict__ enc_proj,
                                                const float* __restrict__ q,
                                                const float* __restrict__ V,
                                                float* __restrict__ scores)
{
    const int wave = threadIdx.x >> 5;
    const int lane = threadIdx.x & 31;
    const int row  = blockIdx.x * 8 + wave;
    const int b    = row >> 11;              // S = 2048
    const float* ep = enc_proj + (size_t)row * 256 + lane * 8;
    const float* qp = q + b * 256 + lane * 8;
    const float* vp = V + lane * 8;

    __builtin_prefetch(ep + 16 * 256, 0, 1); // global_prefetch_b8, stay ahead in L2->L0

    float acc = 0.f;
#pragma unroll
    for (int j = 0; j < 8; ++j)
        acc += fast_tanh(qp[j] + ep[j]) * vp[j];
#pragma unroll
    for (int off = 16; off > 0; off >>= 1)
        acc += __shfl_xor(acc, off, 32);
    if (lane == 0) scores[row] = acc;
}

// ---------------------------------------------------------------------------
// softmax over S=2048 + context[b] = attn @ enc_hiddens[b].
// One block (256 threads) per batch row; attn kept entirely in LDS (8KB).
// ---------------------------------------------------------------------------
__global__ __launch_bounds__(256) void k_softmax_ctx(const float* __restrict__ scores,
                                                     const float* __restrict__ enc,
                                                     float* __restrict__ context)
{
    __shared__ float red[256];
    __shared__ float attn[2048];
    const int b = blockIdx.x, tid = threadIdx.x;
    const float* sc = scores + b * 2048;

    float loc[8];
    float m = -INFINITY;
#pragma unroll
    for (int j = 0; j < 8; ++j) { loc[j] = sc[tid + j * 256]; m = fmaxf(m, loc[j]); }
    red[tid] = m; __syncthreads();
    for (int st = 128; st > 0; st >>= 1) {
        if (tid < st) red[tid] = fmaxf(red[tid], red[tid + st]);
        __syncthreads();
    }
    m = red[0]; __syncthreads();

    float s = 0.f;
#pragma unroll
    for (int j = 0; j < 8; ++j) {
        float e = __expf(loc[j] - m);
        attn[tid + j * 256] = e;
        s += e;
    }
    red[tid] = s; __syncthreads();
    for (int st = 128; st > 0; st >>= 1) {
        if (tid < st) red[tid] += red[tid + st];
        __syncthreads();
    }
    const float inv = 1.f / red[0];
    __syncthreads();
#pragma unroll
    for (int j = 0; j < 8; ++j) attn[tid + j * 256] *= inv;
    __syncthreads();

    // context[b][h], h = tid; each iteration the block reads 4KB contiguous.
    const float* eb = enc + (size_t)b * SENC * 256 + tid;
    float acc = 0.f;
    for (int si = 0; si < SENC; si += 4) {
        acc += attn[si + 0] * eb[(size_t)(si + 0) * 256];
        acc += attn[si + 1] * eb[(size_t)(si + 1) * 256];
        acc += attn[si + 2] * eb[(size_t)(si + 2) * 256];
        acc += attn[si + 3] * eb[(size_t)(si + 3) * 256];
    }
    context[b * 256 + tid] = acc;
}

// ---------------------------------------------------------------------------
// h_new = tanh([x_t|context] @ Wx + hidden @ Wh + bias); also emits out[b][t]
// and next-step q = h_new @ Wq, fused (the block owns the whole h_new row).
// ---------------------------------------------------------------------------
__global__ __launch_bounds__(256) void k_update(const float* __restrict__ dec_seq,
                                                const float* __restrict__ context,
                                                const float* __restrict__ Wx,
                                                const float* __restrict__ Wh,
                                                const float* __restrict__ bias,
                                                const float* __restrict__ W,
                                                int t,
                                                float* __restrict__ hidden,
                                                float* __restrict__ q,
                                                float* __restrict__ out)
{
    __shared__ float inp[768]; // [x_t | context | hidden]
    __shared__ float hn[256];
    const int b = blockIdx.x, h = threadIdx.x;
    inp[h]       = dec_seq[((size_t)b * TDEC + t) * EE + h];
    inp[256 + h] = context[b * 256 + h];
    inp[512 + h] = hidden[b * 256 + h];
    __syncthreads();

    float acc = bias[h];
#pragma unroll 4
    for (int e = 0; e < 512; ++e) acc += inp[e] * Wx[e * 256 + h];
#pragma unroll 4
    for (int k = 0; k < 256; ++k) acc += inp[512 + k] * Wh[k * 256 + h];
    const float hv = fast_tanh(acc);

    hn[h] = hv;
    out[((size_t)b * TDEC + t) * HH + h] = hv;
    hidden[b * 256 + h] = hv;
    __syncthreads();

    float qa = 0.f;
#pragma unroll 4
    for (int k = 0; k < 256; ++k) qa += hn[k] * W[k * 256 + h];
    q[b * 256 + h] = qa;
}

// ---------------------------------------------------------------------------
extern "C" void kernel_launch(void* const* d_in, const int* in_sizes, int n_in,
                              void* d_out, int out_size, void* d_ws, size_t ws_size,
                              hipStream_t stream)
{
    const float* dec_seq = (const float*)d_in[0]; // (B, T, E)
    const float* enc     = (const float*)d_in[1]; // (B, S, H)
    const float* last_h  = (const float*)d_in[2]; // (B, H)
    const float* W       = (const float*)d_in[3]; // (2H, H): Wq rows 0..255, Wk rows 256..511
    const float* V       = (const float*)d_in[4]; // (H,)
    const float* Wx      = (const float*)d_in[5]; // (E+H, H)
    const float* Wh      = (const float*)d_in[6]; // (H, H)
    const float* bias    = (const float*)d_in[7]; // (H,)
    float* out = (float*)d_out;

    // Workspace layout (floats): ~67.4 MB total.
    float* ws       = (float*)d_ws;
    float* enc_proj = ws;                                     // B*S*H = 16,777,216
    float* scores   = enc_proj + (size_t)BB * SENC * HH;      // B*S   = 65,536
    float* context  = scores   + (size_t)BB * SENC;           // B*H
    float* q        = context  + (size_t)BB * HH;             // B*H
    float* hidden   = q        + (size_t)BB * HH;             // B*H

    // One-time fp32 WMMA GEMM: enc_proj = enc_hiddens @ Wk.
    k_enc_proj<<<(BB * SENC) / 16, 256, 0, stream>>>(enc, W, enc_proj);
    // hidden <- last_hidden (never mutate inputs), q <- hidden @ Wq.
    k_init<<<BB, 256, 0, stream>>>(last_h, W, hidden, q);

    for (int t = 0; t < TDEC; ++t) {
        k_scores<<<(BB * SENC) / 8, 256, 0, stream>>>(enc_proj, q, V, scores);
        k_softmax_ctx<<<BB, 256, 0, stream>>>(scores, enc, context);
        k_update<<<BB, 256, 0, stream>>>(dec_seq, context, Wx, Wh, bias, W, t,
                                         hidden, q, out);
    }
}